// LightGCN_Fusion_39960375722249
// MI455X (gfx1250) — compile-verified
//
#include <hip/hip_runtime.h>
#include <hip/hip_bf16.h>

typedef float v2f __attribute__((ext_vector_type(2)));
typedef float v8f __attribute__((ext_vector_type(8)));
typedef int   v4i __attribute__((ext_vector_type(4)));

#define EMBED 64
#define TEXTD 768
#define KCHUNK 192          // K-rows of W staged per stage (4 stages x 192 = 768)
#define WROW 80             // padded LDS row stride in floats (bank-conflict-free)

typedef __attribute__((address_space(1))) v4i g_v4i;   // global b128 unit
typedef __attribute__((address_space(3))) v4i l_v4i;   // LDS b128 unit

// ---------------------------------------------------------------------------
// Copy user embeddings into x0 and acc (float4 / b128 path)
// ---------------------------------------------------------------------------
__global__ void lgcn_init_users(const float* __restrict__ user_emb,
                                float* __restrict__ x,
                                float* __restrict__ acc,
                                int count4) {
  int i = blockIdx.x * blockDim.x + threadIdx.x;
  if (i < count4) {
    float4 v = ((const float4*)user_emb)[i];
    ((float4*)x)[i] = v;
    ((float4*)acc)[i] = v;
  }
}

// ---------------------------------------------------------------------------
// item0 = item_id_emb + text_emb @ W + b   via V_WMMA_F32_16X16X4_F32
// W is staged into LDS (4 stages of 192 K-rows, async global->LDS copy),
// B fragments come from ds_load; A fragments stream from global.
//
// Fragment layouts (ISA 7.12.2, fp32):
//   A 16x4:  lane l<16 -> {A[l][k], A[l][k+1]}, lane l>=16 -> {A[l-16][k+2], A[l-16][k+3]}
//   B 4x16:  vgpr0: lanes0-15 = row k, lanes16-31 = row k+2; vgpr1: rows k+1 / k+3
//   C 16x16: vgpr v: lanes0-15 -> (M=v, N=lane), lanes16-31 -> (M=v+8, N=lane-16)
// ---------------------------------------------------------------------------
__global__ void __launch_bounds__(256)
lgcn_item_proj_wmma(const float* __restrict__ text,    // [ni,768]
                    const float* __restrict__ W,       // [768,64]
                    const float* __restrict__ bias,    // [64]
                    const float* __restrict__ item_id, // [ni,64]
                    float* __restrict__ x_items,       // [ni,64]
                    float* __restrict__ acc_items,     // [ni,64]
                    int num_strips) {
  __shared__ float Wl[KCHUNK * WROW];   // 61,440 B

  const int lane  = threadIdx.x & 31;
  const int strip = blockIdx.x * (blockDim.x >> 5) + (threadIdx.x >> 5);
  const bool active = (strip < num_strips);   // wave-uniform predicate

  const int row0 = strip * 16;
  const int m  = lane & 15;                   // A row / B col within tile
  const int kh = lane >> 4;                   // K-pair select (0 or 1)

  v8f c0 = {}, c1 = {}, c2 = {}, c3 = {};
  const float* arow = active ? (text + (size_t)(row0 + m) * TEXTD + kh * 2) : text;

  for (int s = 0; s < TEXTD / KCHUNK; ++s) {
    const int k0 = s * KCHUNK;

    __syncthreads();  // previous stage's readers done before LDS overwrite

    // ---- stage W[k0 .. k0+191][0..63] into LDS (padded rows) -------------
    // 192 rows * 16 chunks of 16B = 3072 units; 256 threads * 12 units each.
#pragma unroll
    for (int j = 0; j < (KCHUNK * (EMBED / 4)) / 256; ++j) {
      int u   = threadIdx.x + 256 * j;
      int row = u >> 4;          // 0..191
      int c4  = (u & 15) * 4;    // float offset within row
      const float* gsrc = W + (size_t)(k0 + row) * EMBED + c4;
      float*       ldst = Wl + row * WROW + c4;
#if __has_builtin(__builtin_amdgcn_global_load_async_to_lds_b128)
      __builtin_amdgcn_global_load_async_to_lds_b128(
          (g_v4i*)gsrc, (l_v4i*)ldst, 0, 0);
#else
      *(float4*)ldst = *(const float4*)gsrc;
#endif
    }
#if __has_builtin(__builtin_amdgcn_global_load_async_to_lds_b128)
#if __has_builtin(__builtin_amdgcn_s_wait_asynccnt)
    __builtin_amdgcn_s_wait_asynccnt(0);
#else
    asm volatile("s_wait_asynccnt 0" ::: "memory");
#endif
#endif
    __syncthreads();

    // ---- compute over this K chunk ---------------------------------------
    if (active) {
#pragma unroll 4
      for (int k = 0; k < KCHUNK; k += 4) {
        v2f a = *(const v2f*)(arow + k0 + k);   // 8B-aligned global A fragment

        const int idx = (k + kh * 2) * WROW + m;
        v2f b0, b1, b2, b3;
        b0.x = Wl[idx +  0];  b0.y = Wl[idx +  0 + WROW];
        b1.x = Wl[idx + 16];  b1.y = Wl[idx + 16 + WROW];
        b2.x = Wl[idx + 32];  b2.y = Wl[idx + 32 + WROW];
        b3.x = Wl[idx + 48];  b3.y = Wl[idx + 48 + WROW];

        c0 = __builtin_amdgcn_wmma_f32_16x16x4_f32(false, a, false, b0, (short)0, c0, false, false);
        c1 = __builtin_amdgcn_wmma_f32_16x16x4_f32(false, a, false, b1, (short)0, c1, false, false);
        c2 = __builtin_amdgcn_wmma_f32_16x16x4_f32(false, a, false, b2, (short)0, c2, false, false);
        c3 = __builtin_amdgcn_wmma_f32_16x16x4_f32(false, a, false, b3, (short)0, c3, false, false);
      }
    }
  }

  if (!active) return;

  // Epilogue: + item_id_emb + bias, write x0 and acc
  const int half = kh * 8;
  const int nc   = m;
#pragma unroll
  for (int v = 0; v < 8; ++v) {
    const int r = row0 + v + half;
    const size_t base = (size_t)r * EMBED;
    float o0 = c0[v] + item_id[base +  0 + nc] + bias[ 0 + nc];
    float o1 = c1[v] + item_id[base + 16 + nc] + bias[16 + nc];
    float o2 = c2[v] + item_id[base + 32 + nc] + bias[32 + nc];
    float o3 = c3[v] + item_id[base + 48 + nc] + bias[48 + nc];
    x_items[base +  0 + nc] = o0;  acc_items[base +  0 + nc] = o0;
    x_items[base + 16 + nc] = o1;  acc_items[base + 16 + nc] = o1;
    x_items[base + 32 + nc] = o2;  acc_items[base + 32 + nc] = o2;
    x_items[base + 48 + nc] = o3;  acc_items[base + 48 + nc] = o3;
  }
}

// Scalar tail for item rows not covered by 16-row strips (0 rows for ni=50000)
__global__ void lgcn_item_proj_tail(const float* __restrict__ text,
                                    const float* __restrict__ W,
                                    const float* __restrict__ bias,
                                    const float* __restrict__ item_id,
                                    float* __restrict__ x_items,
                                    float* __restrict__ acc_items,
                                    int row_begin, int row_end) {
  int t = blockIdx.x * blockDim.x + threadIdx.x;
  int r = row_begin + t / EMBED;
  int c = t % EMBED;
  if (r >= row_end) return;
  const float* arow = text + (size_t)r * TEXTD;
  float s = 0.f;
  for (int k = 0; k < TEXTD; ++k) s += arow[k] * W[(size_t)k * EMBED + c];
  float o = s + item_id[(size_t)r * EMBED + c] + bias[c];
  x_items[(size_t)r * EMBED + c] = o;
  acc_items[(size_t)r * EMBED + c] = o;
}

// ---------------------------------------------------------------------------
// SpMM scatter: x_next[rows[e]] += vals[e] * x_cur[cols[e]]
// 4 threads per edge, 16 features each; float4 gathers (L2-resident x),
// non-returning global_atomic_add_f32 scatters.
// ---------------------------------------------------------------------------
__global__ void lgcn_spmm_scatter(const int* __restrict__ rows,
                                  const int* __restrict__ cols,
                                  const float* __restrict__ vals,
                                  const float* __restrict__ xin,
                                  float* __restrict__ xout,
                                  int nnz) {
  int tid = blockIdx.x * blockDim.x + threadIdx.x;
  int e = tid >> 2;
  if (e >= nnz) return;
  int q = (tid & 3) * 16;

  int r = rows[e];
  int c = cols[e];
  float v = vals[e];

  const float4* src = (const float4*)(xin + (size_t)c * EMBED + q);
  float* dst = xout + (size_t)r * EMBED + q;

#pragma unroll
  for (int j = 0; j < 4; ++j) {
    float4 t = src[j];
    unsafeAtomicAdd(dst + j * 4 + 0, v * t.x);
    unsafeAtomicAdd(dst + j * 4 + 1, v * t.y);
    unsafeAtomicAdd(dst + j * 4 + 2, v * t.z);
    unsafeAtomicAdd(dst + j * 4 + 3, v * t.w);
  }
}

__global__ void lgcn_zero(float* __restrict__ p, int count4) {
  int i = blockIdx.x * blockDim.x + threadIdx.x;
  if (i < count4) ((float4*)p)[i] = make_float4(0.f, 0.f, 0.f, 0.f);
}

__global__ void lgcn_acc_add(float* __restrict__ acc, const float* __restrict__ x,
                             int count4) {
  int i = blockIdx.x * blockDim.x + threadIdx.x;
  if (i < count4) {
    float4 a = ((float4*)acc)[i];
    float4 b = ((const float4*)x)[i];
    a.x += b.x; a.y += b.y; a.z += b.z; a.w += b.w;
    ((float4*)acc)[i] = a;
  }
}

__global__ void lgcn_finalize(const float* __restrict__ acc, float* __restrict__ out,
                              float scale, int count4) {
  int i = blockIdx.x * blockDim.x + threadIdx.x;
  if (i < count4) {
    float4 a = ((const float4*)acc)[i];
    a.x *= scale; a.y *= scale; a.z *= scale; a.w *= scale;
    ((float4*)out)[i] = a;
  }
}

// ---------------------------------------------------------------------------
extern "C" void kernel_launch(void* const* d_in, const int* in_sizes, int n_in,
                              void* d_out, int out_size, void* d_ws, size_t ws_size,
                              hipStream_t stream) {
  const float* user_emb = (const float*)d_in[0];
  const float* item_id  = (const float*)d_in[1];
  const float* text     = (const float*)d_in[2];
  const float* W        = (const float*)d_in[3];
  const float* bias     = (const float*)d_in[4];
  const int*   rows     = (const int*)d_in[5];
  const int*   cols     = (const int*)d_in[6];
  const float* vals     = (const float*)d_in[7];

  const int num_users = in_sizes[0] / EMBED;
  const int num_items = in_sizes[1] / EMBED;
  const int nnz       = in_sizes[5];
  const int n         = num_users + num_items;
  const size_t nd     = (size_t)n * EMBED;

  float* xa  = (float*)d_ws;      // x_cur
  float* xb  = xa + nd;           // x_next
  float* acc = xb + nd;           // running sum of layers

  const int B = 256;

  // Layer-0 embedding: users (copy) + items (WMMA projection)
  {
    int cu4 = (num_users * EMBED) / 4;
    lgcn_init_users<<<(cu4 + B - 1) / B, B, 0, stream>>>(user_emb, xa, acc, cu4);

    int strips = num_items / 16;
    if (strips > 0) {
      int waves_per_block = B / 32;
      int blocks = (strips + waves_per_block - 1) / waves_per_block;
      lgcn_item_proj_wmma<<<blocks, B, 0, stream>>>(
          text, W, bias, item_id,
          xa + (size_t)num_users * EMBED, acc + (size_t)num_users * EMBED, strips);
    }
    int tail = num_items - strips * 16;
    if (tail > 0) {
      int tthreads = tail * EMBED;
      lgcn_item_proj_tail<<<(tthreads + B - 1) / B, B, 0, stream>>>(
          text, W, bias, item_id,
          xa + (size_t)num_users * EMBED, acc + (size_t)num_users * EMBED,
          strips * 16, num_items);
    }
  }

  // 3 propagation layers: x_next = A x_cur ; acc += x_next
  const int tot4 = (int)(nd / 4);
  for (int layer = 0; layer < 3; ++layer) {
    lgcn_zero<<<(tot4 + B - 1) / B, B, 0, stream>>>(xb, tot4);
    int sthreads = nnz * 4;
    lgcn_spmm_scatter<<<(sthreads + B - 1) / B, B, 0, stream>>>(
        rows, cols, vals, xa, xb, nnz);
    lgcn_acc_add<<<(tot4 + B - 1) / B, B, 0, stream>>>(acc, xb, tot4);
    float* t = xa; xa = xb; xb = t;  // host-side ping-pong (deterministic)
  }

  // final = acc / 4
  lgcn_finalize<<<(tot4 + B - 1) / B, B, 0, stream>>>(acc, (float*)d_out, 0.25f, tot4);
}